// GAT_44435731644446
// MI455X (gfx1250) — compile-verified
//
#include <hip/hip_runtime.h>
#include <hip/hip_bf16.h>
#include <math.h>

typedef float v2f __attribute__((ext_vector_type(2)));
typedef float v8f __attribute__((ext_vector_type(8)));

// ---------------------------------------------------------------------------
// WMMA fp32 GEMM:  D[M x Nc] = A[M x K] * B[K x Nc]   (row-major everywhere)
// Block = 256 threads = 8 waves; block tile 128x64; wave tile 32x32 (2x2 WMMA)
// Double-buffered LDS, pre-swizzled B tile for contiguous b64 fragments.
// ---------------------------------------------------------------------------
#define GEMM_BM 128
#define GEMM_BN 64
#define GEMM_KT 32
#define LDA_PAD 36    // floats; 16B-aligned rows AND bank-conflict-free (36r mod 64)
#define LDB_PAD 160   // floats per k-pair row; 160 mod 64 == 32 -> halves use disjoint banks

__global__ __launch_bounds__(256) void gat_wmma_gemm_f32(
    const float* __restrict__ A, const float* __restrict__ B,
    float* __restrict__ D, int M, int K, int Nc)
{
    __shared__ float As[2][GEMM_BM * LDA_PAD];        // 2 * 18432 B
    __shared__ float Bs[2][(GEMM_KT / 2) * LDB_PAD];  // 2 * 10240 B

    const int tid   = threadIdx.x;
    const int lane  = tid & 31;
    const int wave  = tid >> 5;      // 0..7
    const int wm    = wave & 3;      // wave M position (x32)
    const int wn    = wave >> 2;     // wave N position (x32)
    const int l16   = lane & 15;
    const int lhalf = lane >> 4;     // 0 or 1

    const int tileM = blockIdx.y * GEMM_BM;
    const int tileN = blockIdx.x * GEMM_BN;

    // staging coordinates (constant per thread)
    int aRow[4], aCol[4], bRow[2], bCol[2];
#pragma unroll
    for (int it = 0; it < 4; ++it) {
        int idx = tid + it * 256;          // 0..1023 over 128x(8 float4)
        aRow[it] = idx >> 3;
        aCol[it] = (idx & 7) * 4;
    }
#pragma unroll
    for (int it = 0; it < 2; ++it) {
        int idx = tid + it * 256;          // 0..511 over 32x(16 float4)
        bRow[it] = idx >> 4;
        bCol[it] = (idx & 15) * 4;
    }

    v8f acc[2][2];
    const v8f vzero = {0.f,0.f,0.f,0.f,0.f,0.f,0.f,0.f};
#pragma unroll
    for (int s = 0; s < 2; ++s)
#pragma unroll
        for (int t = 0; t < 2; ++t) acc[s][t] = vzero;

    const int nT = K / GEMM_KT;
    float4 aReg[4], bReg[2];

    // stage K-tile 0
#pragma unroll
    for (int it = 0; it < 4; ++it) {
        int gr = tileM + aRow[it];
        aReg[it] = make_float4(0.f, 0.f, 0.f, 0.f);
        if (gr < M) aReg[it] = *(const float4*)(A + (size_t)gr * K + aCol[it]);
    }
#pragma unroll
    for (int it = 0; it < 2; ++it)
        bReg[it] = *(const float4*)(B + (size_t)bRow[it] * Nc + tileN + bCol[it]);

    for (int kt = 0; kt < nT; ++kt) {
        const int buf = kt & 1;

        // ---- commit staged tile to LDS ----
#pragma unroll
        for (int it = 0; it < 4; ++it)
            *(float4*)&As[buf][aRow[it] * LDA_PAD + aCol[it]] = aReg[it];
#pragma unroll
        for (int it = 0; it < 2; ++it) {
            int kp = bRow[it] >> 1, ko = bRow[it] & 1;
            float* bp = &Bs[buf][kp * LDB_PAD + bCol[it] * 2 + ko];
            bp[0] = bReg[it].x; bp[2] = bReg[it].y;
            bp[4] = bReg[it].z; bp[6] = bReg[it].w;
        }
        __syncthreads();

        // ---- stage next tile while WMMAs run ----
        if (kt + 1 < nT) {
            const int k0 = (kt + 1) * GEMM_KT;
#pragma unroll
            for (int it = 0; it < 4; ++it) {
                int gr = tileM + aRow[it];
                aReg[it] = make_float4(0.f, 0.f, 0.f, 0.f);
                if (gr < M) aReg[it] = *(const float4*)(A + (size_t)gr * K + k0 + aCol[it]);
            }
#pragma unroll
            for (int it = 0; it < 2; ++it)
                bReg[it] = *(const float4*)(B + (size_t)(k0 + bRow[it]) * Nc + tileN + bCol[it]);
        }

        // ---- 8 x (4 WMMA) over the K chunk ----
#pragma unroll
        for (int kk = 0; kk < GEMM_KT; kk += 4) {
            const int kf = kk + 2 * lhalf;     // ISA 16x4 f32 fragment layout
            v2f af[2], bf[2];
#pragma unroll
            for (int s = 0; s < 2; ++s) {
                int r = wm * 32 + s * 16 + l16;
                af[s] = *(const v2f*)&As[buf][r * LDA_PAD + kf];
            }
#pragma unroll
            for (int t = 0; t < 2; ++t) {
                int n = wn * 32 + t * 16 + l16;
                bf[t] = *(const v2f*)&Bs[buf][(kf >> 1) * LDB_PAD + n * 2];
            }
#pragma unroll
            for (int s = 0; s < 2; ++s)
#pragma unroll
                for (int t = 0; t < 2; ++t)
                    acc[s][t] = __builtin_amdgcn_wmma_f32_16x16x4_f32(
                        false, af[s], false, bf[t],
                        (short)0, acc[s][t], false, false);
        }
    }

    // ---- store D: lane l, vgpr v -> row v + 8*(l>>4), col l&15 ----
#pragma unroll
    for (int s = 0; s < 2; ++s) {
#pragma unroll
        for (int t = 0; t < 2; ++t) {
            int n = tileN + wn * 32 + t * 16 + l16;
#pragma unroll
            for (int v = 0; v < 8; ++v) {
                int m = tileM + wm * 32 + s * 16 + v + 8 * lhalf;
                if (m < M)
                    D[(size_t)m * Nc + n] = acc[s][t][v];
            }
        }
    }
}

// ---------------------------------------------------------------------------
// Edge-phase kernels. H=4, C=64, HC=256 for every layer of this model.
// ---------------------------------------------------------------------------
__device__ __forceinline__ unsigned fenc(float f) {
    int i = __float_as_int(f);
    return (i >= 0) ? ((unsigned)i | 0x80000000u) : (unsigned)(~i);
}
__device__ __forceinline__ float fdec(unsigned u) {
    int i = (u & 0x80000000u) ? (int)(u & 0x7fffffffu) : ~(int)u;
    return __int_as_float(i);
}

__global__ __launch_bounds__(256) void gat_fill_f32(float* p, float v, int n) {
    int i = blockIdx.x * 256 + threadIdx.x;
    if (i < n) p[i] = v;
}
__global__ __launch_bounds__(256) void gat_fill_u32(unsigned* p, unsigned v, int n) {
    int i = blockIdx.x * 256 + threadIdx.x;
    if (i < n) p[i] = v;
}

// pass 1: per (edge, head) logit + segment max (float4-vectorized gathers)
__global__ __launch_bounds__(256) void gat_logits(
    const float* __restrict__ xl, const float* __restrict__ xr,
    const int* __restrict__ src, const int* __restrict__ dst,
    const float* __restrict__ att, float* __restrict__ logits,
    unsigned* __restrict__ mbuf, int E, int Etot)
{
    int idx = blockIdx.x * 256 + threadIdx.x;
    if (idx >= Etot * 4) return;
    int e = idx >> 2, h = idx & 3;
    int s, d;
    if (e < E) { s = src[e]; d = dst[e]; } else { s = e - E; d = s; }
    const float4* pl = (const float4*)(xl + (size_t)s * 256 + h * 64);
    const float4* pr = (const float4*)(xr + (size_t)d * 256 + h * 64);
    const float4* pa = (const float4*)(att + h * 64);
    float acc = 0.f;
#pragma unroll
    for (int c = 0; c < 16; ++c) {
        float4 a = pl[c], b = pr[c], w = pa[c];
        float v0 = a.x + b.x, v1 = a.y + b.y, v2 = a.z + b.z, v3 = a.w + b.w;
        v0 = (v0 > 0.f) ? v0 : 0.2f * v0;   // leaky_relu(0.2)
        v1 = (v1 > 0.f) ? v1 : 0.2f * v1;
        v2 = (v2 > 0.f) ? v2 : 0.2f * v2;
        v3 = (v3 > 0.f) ? v3 : 0.2f * v3;
        acc = fmaf(v0, w.x, acc); acc = fmaf(v1, w.y, acc);
        acc = fmaf(v2, w.z, acc); acc = fmaf(v3, w.w, acc);
    }
    logits[idx] = acc;
    atomicMax(mbuf + (size_t)d * 4 + h, fenc(acc));
}

// pass 2: ex = exp(logit - max); segment sum of ex
__global__ __launch_bounds__(256) void gat_expsum(
    const float* __restrict__ logits,
    const int* __restrict__ dst,
    const unsigned* __restrict__ mbuf,
    float* __restrict__ exb, float* __restrict__ denom, int E, int Etot)
{
    int idx = blockIdx.x * 256 + threadIdx.x;
    if (idx >= Etot * 4) return;
    int e = idx >> 2, h = idx & 3;
    int d = (e < E) ? dst[e] : (e - E);
    float m  = fdec(mbuf[(size_t)d * 4 + h]);
    float ex = expf(logits[idx] - m);
    exb[idx] = ex;
    atomicAdd(denom + (size_t)d * 4 + h, ex);
}

// pass 3: out[d] += alpha * xl[src];  1 block per edge, 256 channels
__global__ __launch_bounds__(256) void gat_scatter(
    const float* __restrict__ xl,
    const int* __restrict__ src, const int* __restrict__ dst,
    const float* __restrict__ exb, const float* __restrict__ denom,
    float* __restrict__ acc, int E)
{
    int e = blockIdx.x;
    int j = threadIdx.x;        // 0..255, head = j>>6
    int h = j >> 6;
    int s, d;
    if (e < E) { s = src[e]; d = dst[e]; } else { s = e - E; d = s; }
    float alpha = exb[(size_t)e * 4 + h] / (denom[(size_t)d * 4 + h] + 1e-16f);
    atomicAdd(acc + (size_t)d * 256 + j, alpha * xl[(size_t)s * 256 + j]);
}

// bias + BN(eval) + ReLU
__global__ __launch_bounds__(256) void gat_bn_relu(
    const float* __restrict__ acc, const float* __restrict__ bias,
    const float* __restrict__ g, const float* __restrict__ be,
    const float* __restrict__ mean, const float* __restrict__ var,
    float* __restrict__ out, int total)
{
    int idx = blockIdx.x * 256 + threadIdx.x;
    if (idx >= total) return;
    int j = idx & 255;
    float t = acc[idx] + bias[j];
    float y = (t - mean[j]) * rsqrtf(var[j] + 1e-5f) * g[j] + be[j];
    out[idx] = fmaxf(y, 0.f);
}

// final layer: mean over 4 heads + bias
__global__ __launch_bounds__(256) void gat_head_mean(
    const float* __restrict__ acc, const float* __restrict__ b2,
    float* __restrict__ out, int total)
{
    int idx = blockIdx.x * 256 + threadIdx.x;
    if (idx >= total) return;
    int n = idx >> 6, c = idx & 63;
    const float* p = acc + (size_t)n * 256 + c;
    float s = p[0] + p[64] + p[128] + p[192];
    out[idx] = s * 0.25f + b2[c];
}

// ---------------------------------------------------------------------------
extern "C" void kernel_launch(void* const* d_in, const int* in_sizes, int n_in,
                              void* d_out, int out_size, void* d_ws, size_t ws_size,
                              hipStream_t stream)
{
    const float* x   = (const float*)d_in[0];
    const int*   ei  = (const int*)d_in[1];
    const float* w0l = (const float*)d_in[2];
    const float* w0r = (const float*)d_in[3];
    const float* a0  = (const float*)d_in[4];
    const float* b0  = (const float*)d_in[5];
    const float* g0  = (const float*)d_in[6];
    const float* be0 = (const float*)d_in[7];
    const float* m0  = (const float*)d_in[8];
    const float* v0  = (const float*)d_in[9];
    const float* w1l = (const float*)d_in[10];
    const float* w1r = (const float*)d_in[11];
    const float* a1  = (const float*)d_in[12];
    const float* b1  = (const float*)d_in[13];
    const float* g1  = (const float*)d_in[14];
    const float* be1 = (const float*)d_in[15];
    const float* m1  = (const float*)d_in[16];
    const float* v1  = (const float*)d_in[17];
    const float* w2l = (const float*)d_in[18];
    const float* w2r = (const float*)d_in[19];
    const float* a2  = (const float*)d_in[20];
    const float* b2  = (const float*)d_in[21];
    float* out = (float*)d_out;

    const int N    = in_sizes[0] / 128;   // 50000
    const int E    = in_sizes[1] / 2;     // 600000
    const int Etot = E + N;               // + self loops
    const int* src = ei;                  // edge_index[0]
    const int* dst = ei + E;              // edge_index[1]
    const int HC   = 256;

    // workspace carve-up
    char* w = (char*)d_ws;
    float*    xl     = (float*)w;    w += (size_t)N * HC * sizeof(float);
    float*    xr     = (float*)w;    w += (size_t)N * HC * sizeof(float);
    float*    accbuf = (float*)w;    w += (size_t)N * HC * sizeof(float);
    float*    hbuf   = (float*)w;    w += (size_t)N * HC * sizeof(float);
    float*    logits = (float*)w;    w += (size_t)Etot * 4 * sizeof(float);
    float*    exb    = (float*)w;    w += (size_t)Etot * 4 * sizeof(float);
    unsigned* mbuf   = (unsigned*)w; w += (size_t)N * 4 * sizeof(unsigned);
    float*    denom  = (float*)w;    w += (size_t)N * 4 * sizeof(float);

    const dim3 gemm_grid(HC / GEMM_BN, (N + GEMM_BM - 1) / GEMM_BM);
    const int  eh_blocks  = (Etot * 4 + 255) / 256;
    const int  nhc_blocks = (N * HC + 255) / 256;
    const int  nh_blocks  = (N * 4 + 255) / 256;

    const float* Win  = x;
    int          Kin  = 128;
    const float* Wls[3]  = {w0l, w1l, w2l};
    const float* Wrs[3]  = {w0r, w1r, w2r};
    const float* atts[3] = {a0, a1, a2};

    for (int layer = 0; layer < 3; ++layer) {
        // dense transforms on the matrix pipe
        gat_wmma_gemm_f32<<<gemm_grid, 256, 0, stream>>>(Win, Wls[layer], xl, N, Kin, HC);
        gat_wmma_gemm_f32<<<gemm_grid, 256, 0, stream>>>(Win, Wrs[layer], xr, N, Kin, HC);

        // reset segment buffers
        gat_fill_u32<<<nh_blocks, 256, 0, stream>>>(mbuf, 0u, N * 4);
        gat_fill_f32<<<nh_blocks, 256, 0, stream>>>(denom, 0.f, N * 4);
        gat_fill_f32<<<nhc_blocks, 256, 0, stream>>>(accbuf, 0.f, N * HC);

        // segment softmax + aggregation
        gat_logits <<<eh_blocks, 256, 0, stream>>>(xl, xr, src, dst, atts[layer],
                                                   logits, mbuf, E, Etot);
        gat_expsum <<<eh_blocks, 256, 0, stream>>>(logits, dst, mbuf, exb, denom, E, Etot);
        gat_scatter<<<Etot, 256, 0, stream>>>(xl, src, dst, exb, denom, accbuf, E);

        if (layer < 2) {
            const float* bb[2]  = {b0, b1};
            const float* gg[2]  = {g0, g1};
            const float* bee[2] = {be0, be1};
            const float* mm[2]  = {m0, m1};
            const float* vv[2]  = {v0, v1};
            gat_bn_relu<<<nhc_blocks, 256, 0, stream>>>(accbuf, bb[layer], gg[layer],
                                                        bee[layer], mm[layer], vv[layer],
                                                        hbuf, N * HC);
            Win = hbuf;
            Kin = HC;
        } else {
            gat_head_mean<<<(N * 64 + 255) / 256, 256, 0, stream>>>(accbuf, b2, out, N * 64);
        }
    }
}